// RelativePosition_2714419331682
// MI455X (gfx1250) — compile-verified
//
#include <hip/hip_runtime.h>

typedef __attribute__((ext_vector_type(2))) float v2f;
typedef __attribute__((ext_vector_type(8))) float v8f;

#define NN        1024
#define BATCH     128
#define NT        64                      // 16-wide tiles per dimension
#define TRI       (NN * (NN - 1) / 2)     // 523776 outputs per batch row
#define TILES_PB  (NT * (NT + 1) / 2)     // 2080 upper-tri tiles (incl. diagonal)

// tiles in tile-rows < t:  f(t) = t*(2*NT + 1 - t)/2
__device__ __forceinline__ int tri_prefix(int t) {
    return (t * (2 * NT + 1 - t)) >> 1;
}

__global__ __launch_bounds__(256)
void relpos_wmma_kernel(const float* __restrict__ a, float* __restrict__ out) {
    const int lane = threadIdx.x & 31;

    // Wave-uniform tile id, forced scalar so all tile math lives in SGPRs.
    const int gwave =
        __builtin_amdgcn_readfirstlane((int)(blockIdx.x * 8) + (int)(threadIdx.x >> 5));
    if (gwave >= BATCH * TILES_PB) return;

    const int b = gwave / TILES_PB;
    const int r = gwave - b * TILES_PB;

    // decode tile (ti, tj); fp32 sqrt estimate is within +-1 here, fix in one step
    int ti_e = (int)((129.0f - sqrtf(16641.0f - 8.0f * (float)r)) * 0.5f);
    if (ti_e < 1) ti_e = 1;                     // keep ti_e-1 legal below
    if (ti_e > NT - 1) ti_e = NT - 1;
    if (tri_prefix(ti_e) > r) ti_e -= 1;        // step down if overshot
    if (ti_e < NT - 1 && tri_prefix(ti_e + 1) <= r) ti_e += 1;  // step up if undershot

    const int ti = __builtin_amdgcn_readfirstlane(ti_e);
    const int tj = __builtin_amdgcn_readfirstlane(ti_e + (r - tri_prefix(ti_e)));

    const int i0 = ti << 4;
    const int j0 = tj << 4;

    const float* arow = a + b * NN;             // scalar base
    const int  l15 = lane & 15;
    const bool lo  = lane < 16;

    const float ai = arow[i0 + l15];            // rows i0..i0+15 (lanes 0-15 feed MMA)
    const float aj = arow[j0 + l15];            // cols j0..j0+15

    // A is 16x4 f32: lanes 0-15 hold (K=0,K=1), lanes 16-31 hold (K=2,K=3).
    // Only K=0 is populated -> layout-robust rank-1 products.
    v2f A1; A1.x = lo ? -ai  : 0.0f; A1.y = 0.0f;   // A1[m,0] = -a_i[m]
    v2f B1; B1.x = lo ? 1.0f : 0.0f; B1.y = 0.0f;   // B1[0,n] = 1
    v2f A2; A2.x = lo ? 1.0f : 0.0f; A2.y = 0.0f;   // A2[m,0] = 1
    v2f B2; B2.x = lo ? aj   : 0.0f; B2.y = 0.0f;   // B2[0,n] = a_j[n]

    v8f c = {};
    // c = (-a_i) x 1^T
    c = __builtin_amdgcn_wmma_f32_16x16x4_f32(false, A1, false, B1,
                                              (short)0, c, false, false);
    // c += 1 x a_j^T   ->  c[m,n] = a[j0+n] - a[i0+m]
    c = __builtin_amdgcn_wmma_f32_16x16x4_f32(false, A2, false, B2,
                                              (short)0, c, false, false);

    // D layout: VGPR v, lanes 0-15 -> row m=v; lanes 16-31 -> row m=v+8; col n=lane&15
    const int i_first = i0 + (lo ? 0 : 8);      // row of c[0] for this lane
    const int j       = j0 + l15;

    // packed offset of (i_first, j): rowBase(i) + j - i - 1, rowBase(i)=i*(N-1)-i(i-1)/2
    int off = i_first * (NN - 1) - ((i_first * (i_first - 1)) >> 1) + (j - i_first - 1);
    float* __restrict__ outb = out + (size_t)b * TRI;   // scalar base -> saddr stores

    if (tj > ti) {                               // scalar branch: 97% of waves
#pragma unroll
        for (int v = 0; v < 8; ++v) {
            outb[off] = c[v];
            off += NN - 2 - (i_first + v);       // next row, same column j
        }
    } else {                                     // diagonal tile: per-lane predicate
        int iv = i_first;
#pragma unroll
        for (int v = 0; v < 8; ++v) {
            if (j > iv) outb[off] = c[v];
            off += NN - 2 - iv;
            ++iv;
        }
    }
}

extern "C" void kernel_launch(void* const* d_in, const int* in_sizes, int n_in,
                              void* d_out, int out_size, void* d_ws, size_t ws_size,
                              hipStream_t stream) {
    const float* a  = (const float*)d_in[0];
    float* out      = (float*)d_out;
    const int totalWaves = BATCH * TILES_PB;          // 266,240 wave-tiles
    const int blocks     = (totalWaves + 7) / 8;      // 8 waves per 256-thread block
    relpos_wmma_kernel<<<blocks, 256, 0, stream>>>(a, out);
}